// LocalFeatureEncoder_40063454937486
// MI455X (gfx1250) — compile-verified
//
#include <hip/hip_runtime.h>
#include <hip/hip_bf16.h>

// ---------------------------------------------------------------------------
// LocalFeatureEncoder for MI455X (gfx1250, wave32)
//   Stage 1: pair kernel  -> feats[B*N][36]  (onehot(2) + env(32) + pad(2))
//   Stage 2: MLP kernel   -> 3 GEMM layers via V_WMMA_F32_16X16X4_F32 (f32)
// ---------------------------------------------------------------------------

#define NN      1024
#define RBF     16
#define CUTOFF  2.5f
#define FROW    36          // padded in_dim (34 -> 36, multiple of 4)

typedef __attribute__((ext_vector_type(2))) float v2f;
typedef __attribute__((ext_vector_type(8))) float v8f;

// ---------------------------------------------------------------------------
// Stage 1: pairwise RBF environment features.
// grid.x = B*16 blocks; block = 256 threads = 64 i-atoms x 4 j-chunks.
// Deterministic: fixed-order LDS reduction, no atomics.
// ---------------------------------------------------------------------------
__global__ void __launch_bounds__(256)
lfe_pair_kernel(const float* __restrict__ pos,
                const int*   __restrict__ types,
                float*       __restrict__ feats)
{
    __shared__ __align__(16) float sx[NN];
    __shared__ __align__(16) float sy[NN];
    __shared__ __align__(16) float sz[NN];
    __shared__ __align__(16) int   st[NN];
    __shared__ __align__(16) float red[256 * 32];   // 32 KB partials

    const int b   = blockIdx.x >> 4;     // batch
    const int it  = blockIdx.x & 15;     // i-tile (64 atoms)
    const int tid = threadIdx.x;

    // Stage all N=1024 neighbor positions + types for this batch into LDS.
    const float* pb = pos + (size_t)b * NN * 3;
    for (int j = tid; j < NN; j += 256) {
        sx[j] = pb[j * 3 + 0];
        sy[j] = pb[j * 3 + 1];
        sz[j] = pb[j * 3 + 2];
        st[j] = types[b * NN + j];
    }
    __syncthreads();

    const int il = tid & 63;             // i within tile
    const int js = tid >> 6;             // j-chunk 0..3
    const int i  = it * 64 + il;

    const float xi = sx[i], yi = sy[i], zi = sz[i];

    float a0[RBF], a1[RBF];              // env accumulators per species
#pragma unroll
    for (int r = 0; r < RBF; ++r) { a0[r] = 0.f; a1[r] = 0.f; }

    const float dc     = CUTOFF / 15.f;                       // center spacing
    const float invw2  = 1.0f / (dc * dc + 1e-12f);
    const float invcut = 1.0f / CUTOFF;

    const int j0 = js * 256;
    for (int jj = 0; jj < 256; ++jj) {
        const int j = j0 + jj;
        if (j == i) continue;                                 // diagonal masked
        const float dx = xi - sx[j];
        const float dy = yi - sy[j];
        const float dz = zi - sz[j];
        const float sq = dx * dx + dy * dy + dz * dz;
        const float d  = (sq > 0.f) ? sqrtf(sq) : 0.f;        // safe sqrt
        if (d < CUTOFF) {                                     // ~6.5% of pairs
            const float x  = d * invcut;
            const float x2 = x * x;
            const float x3 = x2 * x;
            const float cutv = 1.0f - x3 * (10.f - 15.f * x + 6.f * x2);
            const int   t  = st[j];
            const float m0 = (t == 0) ? cutv : 0.f;
            const float m1 = cutv - m0;                       // t in {0,1}
#pragma unroll
            for (int r = 0; r < RBF; ++r) {
                const float diff = d - (float)r * dc;
                const float e = __expf(-diff * diff * invw2); // v_exp_f32
                a0[r] += e * m0;
                a1[r] += e * m1;
            }
        }
    }

    // Spill partials, then fixed-order reduce over 4 j-chunks.
    float* myr = &red[tid * 32];
#pragma unroll
    for (int r = 0; r < RBF; ++r) { myr[r] = a0[r]; myr[16 + r] = a1[r]; }
    __syncthreads();

    if (tid < 64) {
        const int row = b * NN + it * 64 + tid;
        float* o = feats + (size_t)row * FROW;
        const int ty = st[it * 64 + tid];
        o[0] = (ty == 0) ? 1.f : 0.f;                         // onehot
        o[1] = (ty == 1) ? 1.f : 0.f;
#pragma unroll
        for (int r = 0; r < RBF; ++r) {
            float s0 = 0.f, s1 = 0.f;
#pragma unroll
            for (int q = 0; q < 4; ++q) {
                s0 += red[(q * 64 + tid) * 32 + r];
                s1 += red[(q * 64 + tid) * 32 + 16 + r];
            }
            o[2 + r * 2 + 0] = s0;                            // env index = r*S + s
            o[2 + r * 2 + 1] = s1;
        }
        o[34] = 0.f;                                          // K padding for WMMA
        o[35] = 0.f;
    }
}

// ---------------------------------------------------------------------------
// Stage 2: 3-layer MLP in full f32 via V_WMMA_F32_16X16X4_F32.
// block = 128 threads = 4 waves; each wave owns a 16-row tile (EXEC all-ones).
// ---------------------------------------------------------------------------
__device__ __forceinline__ float silu(float x) {
    return x / (1.0f + __expf(-x));
}

__global__ void __launch_bounds__(128)
lfe_mlp_kernel(const float* __restrict__ feats,
               const float* __restrict__ W1, const float* __restrict__ b1,
               const float* __restrict__ W2, const float* __restrict__ b2,
               const float* __restrict__ W3, const float* __restrict__ b3,
               float*       __restrict__ out)
{
    __shared__ __align__(16) float hA[4][16 * 64];   // per-wave ping
    __shared__ __align__(16) float hB[4][16 * 64];   // per-wave pong

    const int lane  = threadIdx.x & 31;
    const int wave  = threadIdx.x >> 5;
    const int tile  = blockIdx.x * 4 + wave;
    const int r0    = tile * 16;                     // first row of this tile
    const int m16   = lane & 15;                     // M (A frag) or N (B/C frag)
    const int khalf = lane >> 4;                     // selects K pair {0,1}/{2,3}

    v8f c[4];

    // ---------------- Layer 1: feats[.,36(34)] @ W1[34,64] + b1, SiLU -------
#pragma unroll
    for (int t = 0; t < 4; ++t) {
        const float bv = b1[t * 16 + m16];
        c[t] = (v8f){bv, bv, bv, bv, bv, bv, bv, bv};
    }
    {
        const float* arow = feats + (size_t)(r0 + m16) * FROW + 2 * khalf;
#pragma unroll
        for (int kk = 0; kk < 9; ++kk) {             // K = 36 (padded)
            const v2f a = *(const v2f*)(arow + 4 * kk);
            const int k0 = 4 * kk + 2 * khalf;
#pragma unroll
            for (int t = 0; t < 4; ++t) {
                const int n = t * 16 + m16;
                v2f bb;
                bb.x = (k0     < 34) ? W1[k0 * 64 + n]       : 0.f;
                bb.y = (k0 + 1 < 34) ? W1[(k0 + 1) * 64 + n] : 0.f;
                c[t] = __builtin_amdgcn_wmma_f32_16x16x4_f32(
                           false, a, false, bb, (short)0, c[t], false, false);
            }
        }
    }
    {   // SiLU, C/D layout -> LDS row-major tile
        float* h = hA[wave];
#pragma unroll
        for (int t = 0; t < 4; ++t)
#pragma unroll
            for (int v = 0; v < 8; ++v) {
                const int m = v + 8 * khalf;
                h[m * 64 + t * 16 + m16] = silu(c[t][v]);
            }
    }
    __syncthreads();

    // ---------------- Layer 2: h[.,64] @ W2[64,64] + b2, SiLU ---------------
#pragma unroll
    for (int t = 0; t < 4; ++t) {
        const float bv = b2[t * 16 + m16];
        c[t] = (v8f){bv, bv, bv, bv, bv, bv, bv, bv};
    }
    {
        const float* h = hA[wave];
#pragma unroll
        for (int kk = 0; kk < 16; ++kk) {            // K = 64
            const int k0 = 4 * kk + 2 * khalf;
            const v2f a = *(const v2f*)(&h[m16 * 64 + k0]);
#pragma unroll
            for (int t = 0; t < 4; ++t) {
                const int n = t * 16 + m16;
                v2f bb;
                bb.x = W2[k0 * 64 + n];
                bb.y = W2[(k0 + 1) * 64 + n];
                c[t] = __builtin_amdgcn_wmma_f32_16x16x4_f32(
                           false, a, false, bb, (short)0, c[t], false, false);
            }
        }
    }
    {
        float* h2 = hB[wave];
#pragma unroll
        for (int t = 0; t < 4; ++t)
#pragma unroll
            for (int v = 0; v < 8; ++v) {
                const int m = v + 8 * khalf;
                h2[m * 64 + t * 16 + m16] = silu(c[t][v]);
            }
    }
    __syncthreads();

    // ---------------- Layer 3: h2[.,64] @ W3[64,64] + b3 --------------------
#pragma unroll
    for (int t = 0; t < 4; ++t) {
        const float bv = b3[t * 16 + m16];
        c[t] = (v8f){bv, bv, bv, bv, bv, bv, bv, bv};
    }
    {
        const float* h2 = hB[wave];
#pragma unroll
        for (int kk = 0; kk < 16; ++kk) {            // K = 64
            const int k0 = 4 * kk + 2 * khalf;
            const v2f a = *(const v2f*)(&h2[m16 * 64 + k0]);
#pragma unroll
            for (int t = 0; t < 4; ++t) {
                const int n = t * 16 + m16;
                v2f bb;
                bb.x = W3[k0 * 64 + n];
                bb.y = W3[(k0 + 1) * 64 + n];
                c[t] = __builtin_amdgcn_wmma_f32_16x16x4_f32(
                           false, a, false, bb, (short)0, c[t], false, false);
            }
        }
    }
    // Store D tiles to global output [rows][64].
#pragma unroll
    for (int t = 0; t < 4; ++t)
#pragma unroll
        for (int v = 0; v < 8; ++v) {
            const int m = v + 8 * khalf;
            out[(size_t)(r0 + m) * 64 + t * 16 + m16] = c[t][v];
        }
}

// ---------------------------------------------------------------------------
extern "C" void kernel_launch(void* const* d_in, const int* in_sizes, int n_in,
                              void* d_out, int out_size, void* d_ws, size_t ws_size,
                              hipStream_t stream)
{
    const float* pos   = (const float*)d_in[0];   // [B,N,3]
    const int*   types = (const int*)  d_in[1];   // [B,N]
    const float* W1    = (const float*)d_in[2];   // [34,64]
    const float* b1    = (const float*)d_in[3];   // [64]
    const float* W2    = (const float*)d_in[4];   // [64,64]
    const float* b2    = (const float*)d_in[5];   // [64]
    const float* W3    = (const float*)d_in[6];   // [64,64]
    const float* b3    = (const float*)d_in[7];   // [64]
    float*       out   = (float*)d_out;           // [B,N,64]

    const int B = in_sizes[0] / (NN * 3);         // 8
    float* feats = (float*)d_ws;                  // [B*N][36] padded features

    // Stage 1: pairwise env features (writes every feats element -> no init).
    lfe_pair_kernel<<<B * 16, 256, 0, stream>>>(pos, types, feats);

    // Stage 2: WMMA f32 MLP. One 16-row tile per wave, 4 waves per block.
    const int nrows = B * NN;                     // 8192
    lfe_mlp_kernel<<<nrows / 64, 128, 0, stream>>>(feats, W1, b1, W2, b2,
                                                   W3, b3, out);
}